// GaussianUpsamplingModule_47476568490676
// MI455X (gfx1250) — compile-verified
//
#include <hip/hip_runtime.h>
#include <hip/hip_bf16.h>
#include <math.h>

#define B_  32
#define L_  512
#define C_  256

typedef __attribute__((ext_vector_type(2))) float v2f;
typedef __attribute__((ext_vector_type(8))) float v8f;

// ---------------------------------------------------------------------------
// Kernel 1: convs + xc + projection -> ranges.  One block per (b,l), 256 thr.
// ---------------------------------------------------------------------------
__global__ __launch_bounds__(256) void gu_prep(
    const float* __restrict__ x, const float* __restrict__ durf,
    const float* __restrict__ ener, const float* __restrict__ pit,
    const float* __restrict__ Wd, const float* __restrict__ bd,
    const float* __restrict__ We, const float* __restrict__ be,
    const float* __restrict__ Wp, const float* __restrict__ bp,
    const float* __restrict__ Wproj, const float* __restrict__ bproj,
    const int* __restrict__ ilen,
    float* __restrict__ xc, float* __restrict__ ranges)
{
    const int bl = blockIdx.x;
    const int b  = bl / L_;
    const int l  = bl % L_;
    const int c  = threadIdx.x;      // 0..255 == channel

    const int sb = b * L_;
    const float dm1 = (l > 0)      ? durf[sb + l - 1] : 0.f;
    const float d0  =                durf[sb + l];
    const float dp1 = (l < L_ - 1) ? durf[sb + l + 1] : 0.f;
    const float em1 = (l > 0)      ? ener[sb + l - 1] : 0.f;
    const float e0  =                ener[sb + l];
    const float ep1 = (l < L_ - 1) ? ener[sb + l + 1] : 0.f;
    const float pm1 = (l > 0)      ? pit[sb + l - 1]  : 0.f;
    const float p0  =                pit[sb + l];
    const float pp1 = (l < L_ - 1) ? pit[sb + l + 1]  : 0.f;

    // cross-correlation (JAX conv): y[l] = w0*s[l-1] + w1*s[l] + w2*s[l+1]
    const float dconv = Wd[c*3+0]*dm1 + Wd[c*3+1]*d0 + Wd[c*3+2]*dp1 + bd[c];
    const float econv = We[c*3+0]*em1 + We[c*3+1]*e0 + We[c*3+2]*ep1 + be[c];
    const float pconv = Wp[c*3+0]*pm1 + Wp[c*3+1]*p0 + Wp[c*3+2]*pp1 + bp[c];

    const float xv  = x[((size_t)sb + l) * C_ + c];
    const float xcv = xv + econv + pconv;
    xc[((size_t)sb + l) * C_ + c] = xcv;

    __shared__ float red[256];
    red[c] = (xcv + dconv) * Wproj[c];
    __syncthreads();
    for (int s = 128; s > 0; s >>= 1) {
        if (c < s) red[c] += red[c + s];
        __syncthreads();
    }
    if (c == 0) {
        const float v  = red[0] + bproj[0];
        const float sp = (v > 20.f) ? v : log1pf(__expf(v));
        ranges[sb + l] = (l >= ilen[b]) ? 1.0f : sp;
    }
}

// ---------------------------------------------------------------------------
// Kernel 2: per-batch exclusive cumsum -> means.  One block per batch.
// ---------------------------------------------------------------------------
__global__ void gu_means(const int* __restrict__ duri, float* __restrict__ means)
{
    const int b = blockIdx.x;
    if (threadIdx.x == 0) {
        float run = 0.f;
        for (int l = 0; l < L_; ++l) {
            const int d = duri[b * L_ + l];
            means[b * L_ + l] = 0.5f * (float)d + run;
            run += (float)d;
        }
    }
}

// ---------------------------------------------------------------------------
// Kernel 3: sums[b,t] = sum_l probs(b,l,t).  One block per (b,t), 256 thr.
// ---------------------------------------------------------------------------
__global__ __launch_bounds__(256) void gu_sums(
    const float* __restrict__ means, const float* __restrict__ ranges,
    const int* __restrict__ ilen, float* __restrict__ sums, int T)
{
    const int bt = blockIdx.x;
    const int b  = bt / T;
    const int t  = bt % T;
    const float ft  = (float)t + 0.5f;
    const int   len = ilen[b];
    const int   c   = threadIdx.x;

    float s = 0.f;
    for (int l = c; l < L_; l += 256) {
        if (l < len) {
            const float r    = ranges[b * L_ + l];
            const float invr = 1.f / r;
            const float z    = (ft - means[b * L_ + l]) * invr;
            s += 0.3989422804014327f * __expf(-0.5f * z * z) * invr;
        }
    }
    __shared__ float red[256];
    red[c] = s;
    __syncthreads();
    for (int st = 128; st > 0; st >>= 1) {
        if (c < st) red[c] += red[c + st];
        __syncthreads();
    }
    if (c == 0) sums[(size_t)b * T + t] = red[0];
}

// ---------------------------------------------------------------------------
// Kernel 4 (fused): weights materialization + per-batch GEMM (TxL)*(LxC)
// using V_WMMA_F32_16X16X4_F32.
//
// Block = 512 threads (16 waves).  Block tile: 64 T-rows x 256 C-cols (full C),
// looping over all of L => every (b,l,t) weight belongs to exactly one block.
// The A tile (weights) is therefore COMPUTED during staging (one exp per
// element), stored to LDS for the WMMAs, and streamed once to the weights
// output region.  This removes the 150 MB weights re-read entirely.
// Wave (tRow,cGrp) owns a 16x64 strip => 4 v8f accumulators.
// ---------------------------------------------------------------------------
__global__ __launch_bounds__(512) void gu_einsum_wmma(
    const float* __restrict__ xc,
    const float* __restrict__ means, const float* __restrict__ ranges,
    const int* __restrict__ ilen, const float* __restrict__ sums,
    float* __restrict__ wout, float* __restrict__ out, int T, int nTB)
{
    const int blk = blockIdx.x;
    const int b   = blk / nTB;
    const int tb  = blk % nTB;
    const int t0  = tb * 64;

    __shared__ float As[16][64 + 4];    // [l_local][t_local]  (weights tile)
    __shared__ float Bs[16][256 + 4];   // [l_local][c]        (xc tile)
    __shared__ float meanS[L_];
    __shared__ float invrS[L_];
    __shared__ float wscaleS[L_];       // 0.3989*invr, or 0 on padded rows
    __shared__ float rsumS[64];         // 1/(sums+1e-20), or 0 for t >= T

    const float* Xb = xc + (size_t)b * L_ * C_;
    const int    tid = threadIdx.x;
    const int    len = ilen[b];

    // ---- per-block preload of the cheap (B,L)/(B,T) parameters ----
    {
        const int i = tid;               // 512 threads == L_
        const float m = means[b * L_ + i];
        const float r = ranges[b * L_ + i];
        const float invr = 1.f / r;
        meanS[i]   = m;
        invrS[i]   = invr;
        wscaleS[i] = (i < len) ? 0.3989422804014327f * invr : 0.f;
    }
    if (tid < 64) {
        const int tg = t0 + tid;
        rsumS[tid] = (tg < T) ? 1.f / (sums[(size_t)b * T + tg] + 1e-20f) : 0.f;
    }
    __syncthreads();

    const int wave  = tid >> 5;
    const int lane  = tid & 31;
    const int half  = lane >> 4;   // 0: K pair {0,1}, 1: K pair {2,3}
    const int lid   = lane & 15;
    const int tRow  = wave >> 2;   // 0..3
    const int cGrp  = wave & 3;    // 0..3
    const int tbase = tRow * 16;
    const int cbase = cGrp * 64;

    v8f acc[4];
    #pragma unroll
    for (int j = 0; j < 4; ++j) acc[j] = (v8f){0.f,0.f,0.f,0.f,0.f,0.f,0.f,0.f};

    for (int l0 = 0; l0 < L_; l0 += 16) {
        // ---- stage A tile: COMPUTE weights[b, l0..l0+15, t0..t0+63] ----
        for (int i = tid; i < 16 * 64; i += 512) {
            const int tt = i & 63;
            const int ll = i >> 6;
            const int l  = l0 + ll;
            const int tg = t0 + tt;
            const float ft = (float)tg + 0.5f;
            const float z  = (ft - meanS[l]) * invrS[l];
            const float w  = __expf(-0.5f * z * z) * wscaleS[l] * rsumS[tt];
            As[ll][tt] = w;                                   // 0 beyond T
            if (tg < T)
                wout[((size_t)b * L_ + l) * (size_t)T + tg] = w;
        }
        // ---- stage B tile: xc[b, l0..l0+15, 0..255]  (contiguous in c) ----
        for (int i = tid; i < 16 * 256; i += 512) {
            const int cc = i & 255;
            const int ll = i >> 8;
            Bs[ll][cc] = Xb[(size_t)(l0 + ll) * C_ + cc];
        }
        __syncthreads();

        #pragma unroll
        for (int kk = 0; kk < 16; kk += 4) {
            v2f a;
            a.x = As[kk + 2*half + 0][tbase + lid];
            a.y = As[kk + 2*half + 1][tbase + lid];
            #pragma unroll
            for (int j = 0; j < 4; ++j) {
                v2f bb;
                bb.x = Bs[kk + 2*half + 0][cbase + j*16 + lid];
                bb.y = Bs[kk + 2*half + 1][cbase + j*16 + lid];
                acc[j] = __builtin_amdgcn_wmma_f32_16x16x4_f32(
                    /*neg_a=*/false, a, /*neg_b=*/false, bb,
                    /*c_mod=*/(short)0, acc[j],
                    /*reuse_a=*/false, /*reuse_b=*/false);
            }
        }
        __syncthreads();
    }

    // D layout: VGPR v -> M = v + 8*half, N = lid
    #pragma unroll
    for (int j = 0; j < 4; ++j) {
        #pragma unroll
        for (int v = 0; v < 8; ++v) {
            const int m  = v + 8 * half;
            const int tg = t0 + tbase + m;
            if (tg < T)
                out[((size_t)b * T + tg) * C_ + cbase + j*16 + lid] = acc[j][v];
        }
    }
}

// ---------------------------------------------------------------------------
extern "C" void kernel_launch(void* const* d_in, const int* in_sizes, int n_in,
                              void* d_out, int out_size, void* d_ws, size_t ws_size,
                              hipStream_t stream) {
    const float* x      = (const float*)d_in[0];
    const float* durf   = (const float*)d_in[1];
    const float* ener   = (const float*)d_in[2];
    const float* pit    = (const float*)d_in[3];
    const float* Wd     = (const float*)d_in[4];
    const float* bd     = (const float*)d_in[5];
    const float* We     = (const float*)d_in[6];
    const float* be     = (const float*)d_in[7];
    const float* Wp     = (const float*)d_in[8];
    const float* bp     = (const float*)d_in[9];
    const float* Wproj  = (const float*)d_in[10];
    const float* bproj  = (const float*)d_in[11];
    const int*   duri   = (const int*)d_in[12];
    const int*   ilen   = (const int*)d_in[13];

    // out = [x_upsamp (B,T,C)] ++ [weights (B,L,T)] => out_size = B*T*(C+L)
    const int T = out_size / (B_ * (C_ + L_));

    float* out_up = (float*)d_out;                          // B*T*C
    float* out_w  = (float*)d_out + (size_t)B_ * T * C_;    // B*L*T

    float* xc     = (float*)d_ws;                           // B*L*C
    float* means  = xc + (size_t)B_ * L_ * C_;              // B*L
    float* ranges = means + B_ * L_;                        // B*L
    float* sums   = ranges + B_ * L_;                       // B*T

    gu_prep<<<B_ * L_, 256, 0, stream>>>(x, durf, ener, pit, Wd, bd, We, be,
                                         Wp, bp, Wproj, bproj, ilen, xc, ranges);
    gu_means<<<B_, 32, 0, stream>>>(duri, means);
    gu_sums<<<B_ * T, 256, 0, stream>>>(means, ranges, ilen, sums, T);

    const int nTB = (T + 63) / 64;
    gu_einsum_wmma<<<B_ * nTB, 512, 0, stream>>>(xc, means, ranges, ilen, sums,
                                                 out_w, out_up, T, nTB);
}